// RoIProcessingNetwork_12154757447762
// MI455X (gfx1250) — compile-verified
//
#include <hip/hip_runtime.h>
#include <hip/hip_bf16.h>

typedef float v2f __attribute__((ext_vector_type(2)));
typedef float v8f __attribute__((ext_vector_type(8)));
typedef int   v4i __attribute__((vector_size(4 * sizeof(int))));

#define N_PROP 1000
#define M_PAD  1024
#define NC     80
#define FEAT   12544      // 256*7*7
#define CLIN   1024
#define NCLS_P 128        // 81 padded to 128 (multiple of 64)
#define NBOX   80000      // N_PROP*NC
#define SORTN  131072     // next pow2 >= NBOX
#define K_NMS  1000
#define DWH_CLAMP 4.135166556742356f

// ---- gfx1250 async global->LDS (signature learned from clang diagnostic) ----
#if __has_builtin(__builtin_amdgcn_global_load_async_to_lds_b128)
#define HAVE_ASYNC_LDS 1
#define ASYNC_COPY16(G, L)                                                    \
    __builtin_amdgcn_global_load_async_to_lds_b128(                           \
        (__attribute__((address_space(1))) v4i*)(G),                          \
        (__attribute__((address_space(3))) v4i*)(L), 0, 0)
#else
#define HAVE_ASYNC_LDS 0
#endif

#if __has_builtin(__builtin_amdgcn_s_wait_asynccnt)
#define WAIT_ASYNC(n) __builtin_amdgcn_s_wait_asynccnt(n)
#else
#define WAIT_ASYNC(n) asm volatile("s_wait_asynccnt %0" ::"n"(n) : "memory")
#endif

// ---------------- helpers ----------------
__device__ __forceinline__ unsigned ford(float f) {
    unsigned u = __float_as_uint(f);
    return (u & 0x80000000u) ? ~u : (u | 0x80000000u);
}
__device__ __forceinline__ float funord(unsigned o) {
    unsigned u = (o & 0x80000000u) ? (o ^ 0x80000000u) : ~o;
    return __uint_as_float(u);
}

// ---------------- init: zero pads + pad cls weights ----------------
__global__ void init_ws_kernel(float* A0, unsigned long long* keys,
                               const float* __restrict__ W_cls,
                               const float* __restrict__ b_cls,
                               float* __restrict__ Wc_pad, float* __restrict__ bc_pad) {
    int i = blockIdx.x * blockDim.x + threadIdx.x;
    if (i < (M_PAD - N_PROP) * FEAT)
        A0[(size_t)N_PROP * FEAT + i] = 0.0f;
    if (i < SORTN - NBOX)
        keys[NBOX + i] = 0ULL;
    if (i < CLIN * NCLS_P) {
        int k = i >> 7, n = i & (NCLS_P - 1);
        Wc_pad[i] = (n < 81) ? W_cls[k * 81 + n] : 0.0f;
    }
    if (i < NCLS_P)
        bc_pad[i] = (i < 81) ? b_cls[i] : 0.0f;
}

// ---------------- ROI align (multi-level) ----------------
__global__ void __launch_bounds__(256)
roi_align_kernel(const float* __restrict__ P, const int* __restrict__ imidx,
                 const float* __restrict__ fm0, const float* __restrict__ fm1,
                 const float* __restrict__ fm2, const float* __restrict__ fm3,
                 float* __restrict__ A0) {
    const int n = blockIdx.x;      // proposal
    const int c = threadIdx.x;     // channel
    float x1 = P[n * 4 + 0], y1 = P[n * 4 + 1];
    float x2 = P[n * 4 + 2], y2 = P[n * 4 + 3];
    float w = x2 - x1, h = y2 - y1;
    float lvlf = floorf(4.0f + log2f(sqrtf(fmaxf(w * h, 1e-6f)) / 224.0f));
    int lvl = (int)fminf(fmaxf(lvlf, 2.0f), 5.0f) - 2;

    const float* fm; int H, W; float s;
    if      (lvl == 0) { fm = fm0; H = 200; W = 200; s = 4.0f;  }
    else if (lvl == 1) { fm = fm1; H = 100; W = 100; s = 8.0f;  }
    else if (lvl == 2) { fm = fm2; H = 50;  W = 50;  s = 16.0f; }
    else               { fm = fm3; H = 25;  W = 25;  s = 32.0f; }

    const int b = imidx[n];
    const float* plane = fm + ((size_t)b * 256 + c) * (size_t)H * W;

    __shared__ float sgx[14], sgy[14];
    if (threadIdx.x < 14) {
        float f = (threadIdx.x + 0.5f) / 14.0f;
        float fx1 = x1 / s, fx2 = x2 / s, fy1 = y1 / s, fy2 = y2 / s;
        sgx[threadIdx.x] = fminf(fmaxf(fx1 + f * (fx2 - fx1), 0.0f), (float)(W - 1));
        sgy[threadIdx.x] = fminf(fmaxf(fy1 + f * (fy2 - fy1), 0.0f), (float)(H - 1));
    }
    __syncthreads();

    for (int py = 0; py < 7; ++py) {
        for (int px = 0; px < 7; ++px) {
            float acc = 0.0f;
            #pragma unroll
            for (int sy = 0; sy < 2; ++sy) {
                #pragma unroll
                for (int sx = 0; sx < 2; ++sx) {
                    float gyv = sgy[py * 2 + sy];
                    float gxv = sgx[px * 2 + sx];
                    int y0 = (int)floorf(gyv);
                    int x0 = (int)floorf(gxv);
                    int y1i = min(y0 + 1, H - 1);
                    int x1i = min(x0 + 1, W - 1);
                    float ly = gyv - (float)y0, lx = gxv - (float)x0;
                    float v00 = plane[y0  * W + x0 ];
                    float v01 = plane[y0  * W + x1i];
                    float v10 = plane[y1i * W + x0 ];
                    float v11 = plane[y1i * W + x1i];
                    acc += (1.0f - ly) * ((1.0f - lx) * v00 + lx * v01)
                         +          ly * ((1.0f - lx) * v10 + lx * v11);
                }
            }
            A0[(size_t)n * FEAT + c * 49 + py * 7 + px] = acc * 0.25f;
        }
    }
}

// ---------------- WMMA FP32 GEMM with async double-buffered LDS staging ----
// C[1024 x N] = act(A[1024 x K] * B[K x N] + bias),  N % 64 == 0, K % 64 == 0
__global__ void __launch_bounds__(256)
gemm_wmma_f32(const float* __restrict__ A, const float* __restrict__ B,
              const float* __restrict__ bias, float* __restrict__ C,
              int N, int K, int relu) {
    __shared__ float As[2][64 * 68];
    __shared__ float Bs[2][64 * 68];
    const int tid  = threadIdx.x;
    const int lane = tid & 31;
    const int wave = tid >> 5;            // 0..7
    const int lrow = lane & 15;
    const int half = lane >> 4;
    const int sm   = (wave >> 1) * 16;    // subtile row band (shared by wave's pair)
    const int sn0  = ((wave * 2) & 3) * 16;
    const int sn1  = sn0 + 16;
    const int rowBase = blockIdx.y * 64;
    const int colBase = blockIdx.x * 64;

    v8f acc0 = {0, 0, 0, 0, 0, 0, 0, 0};
    v8f acc1 = {0, 0, 0, 0, 0, 0, 0, 0};

    const int nch = K >> 6;

#if HAVE_ASYNC_LDS
    // issue one 64x64 A chunk + 64x64 B chunk into buffer `buf` (8 async b128/wave)
    auto issue = [&](int kb, int buf) {
        #pragma unroll
        for (int i = 0; i < 4; ++i) {
            int lin = tid + i * 256;      // float4 index, 1024 per matrix
            int r   = lin >> 4;
            int c4  = (lin & 15) * 4;
            ASYNC_COPY16(A + (size_t)(rowBase + r) * K + kb + c4, &As[buf][r * 68 + c4]);
            ASYNC_COPY16(B + (size_t)(kb + r) * N + colBase + c4, &Bs[buf][r * 68 + c4]);
        }
    };
    issue(0, 0);
    for (int ic = 0; ic < nch; ++ic) {
        const int buf = ic & 1;
        const bool pf = (ic + 1) < nch;
        if (pf) issue((ic + 1) << 6, buf ^ 1);
        if (pf) WAIT_ASYNC(8); else WAIT_ASYNC(0);
        __syncthreads();
        const float* as = As[buf];
        const float* bs = Bs[buf];
        #pragma unroll
        for (int kk = 0; kk < 16; ++kk) {
            int k0 = kk * 4 + half * 2;
            v2f a;  a.x  = as[(sm + lrow) * 68 + k0];
                    a.y  = as[(sm + lrow) * 68 + k0 + 1];
            v2f b0; b0.x = bs[(k0    ) * 68 + sn0 + lrow];
                    b0.y = bs[(k0 + 1) * 68 + sn0 + lrow];
            v2f b1; b1.x = bs[(k0    ) * 68 + sn1 + lrow];
                    b1.y = bs[(k0 + 1) * 68 + sn1 + lrow];
            acc0 = __builtin_amdgcn_wmma_f32_16x16x4_f32(false, a, false, b0, (short)0, acc0, false, false);
            acc1 = __builtin_amdgcn_wmma_f32_16x16x4_f32(false, a, false, b1, (short)0, acc1, false, false);
        }
        __syncthreads();
    }
#else
    for (int ic = 0; ic < nch; ++ic) {
        const int kb = ic << 6;
        #pragma unroll
        for (int i = 0; i < 4; ++i) {
            int lin = tid + i * 256;
            int r   = lin >> 4;
            int c4  = (lin & 15) * 4;
            float4 av = *(const float4*)(A + (size_t)(rowBase + r) * K + kb + c4);
            float4 bv = *(const float4*)(B + (size_t)(kb + r) * N + colBase + c4);
            *(float4*)(&As[0][r * 68 + c4]) = av;
            *(float4*)(&Bs[0][r * 68 + c4]) = bv;
        }
        __syncthreads();
        #pragma unroll
        for (int kk = 0; kk < 16; ++kk) {
            int k0 = kk * 4 + half * 2;
            v2f a;  a.x  = As[0][(sm + lrow) * 68 + k0];
                    a.y  = As[0][(sm + lrow) * 68 + k0 + 1];
            v2f b0; b0.x = Bs[0][(k0    ) * 68 + sn0 + lrow];
                    b0.y = Bs[0][(k0 + 1) * 68 + sn0 + lrow];
            v2f b1; b1.x = Bs[0][(k0    ) * 68 + sn1 + lrow];
                    b1.y = Bs[0][(k0 + 1) * 68 + sn1 + lrow];
            acc0 = __builtin_amdgcn_wmma_f32_16x16x4_f32(false, a, false, b0, (short)0, acc0, false, false);
            acc1 = __builtin_amdgcn_wmma_f32_16x16x4_f32(false, a, false, b1, (short)0, acc1, false, false);
        }
        __syncthreads();
    }
#endif

    int gcol0 = colBase + sn0 + lrow;
    int gcol1 = colBase + sn1 + lrow;
    float bv0 = bias[gcol0];
    float bv1 = bias[gcol1];
    #pragma unroll
    for (int r = 0; r < 8; ++r) {
        int grow = rowBase + sm + r + 8 * half;
        float v0 = acc0[r] + bv0;
        float v1 = acc1[r] + bv1;
        if (relu) { v0 = fmaxf(v0, 0.0f); v1 = fmaxf(v1, 0.0f); }
        C[(size_t)grow * N + gcol0] = v0;
        C[(size_t)grow * N + gcol1] = v1;
    }
}

// ---------------- softmax + box decode + key build ----------------
__global__ void decode_kernel(const float* __restrict__ LOGB, const float* __restrict__ REGB,
                              const float* __restrict__ P, float* __restrict__ BX,
                              unsigned long long* __restrict__ keys) {
    int p = blockIdx.x * blockDim.x + threadIdx.x;
    if (p >= N_PROP) return;
    const float* l = LOGB + (size_t)p * NCLS_P;
    float mx = -1e30f;
    for (int j = 0; j < 81; ++j) mx = fmaxf(mx, l[j]);
    float sum = 0.0f;
    for (int j = 0; j < 81; ++j) sum += expf(l[j] - mx);
    float inv = 1.0f / sum;

    float x1 = P[p * 4 + 0], y1 = P[p * 4 + 1];
    float x2 = P[p * 4 + 2], y2 = P[p * 4 + 3];
    float pw = x2 - x1, ph = y2 - y1;
    float pcx = x1 + 0.5f * pw, pcy = y1 + 0.5f * ph;

    for (int c = 0; c < NC; ++c) {
        float sc = expf(l[c] - mx) * inv;
        const float* rg = REGB + (size_t)p * 320 + c * 4;
        float dx = rg[0] * 0.1f, dy = rg[1] * 0.1f;
        float dw = fminf(rg[2] * 0.2f, DWH_CLAMP);
        float dh = fminf(rg[3] * 0.2f, DWH_CLAMP);
        float cx = pcx + dx * pw, cy = pcy + dy * ph;
        float bw = pw * expf(dw), bh = ph * expf(dh);
        float nx1 = fminf(fmaxf(cx - 0.5f * bw, 0.0f), 800.0f);
        float ny1 = fminf(fmaxf(cy - 0.5f * bh, 0.0f), 800.0f);
        float nx2 = fminf(fmaxf(cx + 0.5f * bw, 0.0f), 800.0f);
        float ny2 = fminf(fmaxf(cy + 0.5f * bh, 0.0f), 800.0f);
        int idx = p * NC + c;
        BX[idx * 4 + 0] = nx1; BX[idx * 4 + 1] = ny1;
        BX[idx * 4 + 2] = nx2; BX[idx * 4 + 3] = ny2;
        bool valid = (sc > 0.05f) && ((nx2 - nx1) >= 0.01f) && ((ny2 - ny1) >= 0.01f);
        float ms = valid ? sc : -1.0f;
        keys[idx] = ((unsigned long long)ford(ms) << 32) |
                    (unsigned long long)(0x7FFFFFFFu - (unsigned)idx);
    }
}

// ---------------- global bitonic sort step (descending) ----------------
__global__ void bitonic_step_kernel(unsigned long long* keys, int k, int j) {
    int i = blockIdx.x * blockDim.x + threadIdx.x;
    int ixj = i ^ j;
    if (ixj > i) {
        unsigned long long a = keys[i], b = keys[ixj];
        bool up = ((i & k) == 0);
        if (up ? (a < b) : (a > b)) { keys[i] = b; keys[ixj] = a; }
    }
}

// ---------------- gather top-K_NMS ----------------
__global__ void gather_top_kernel(const unsigned long long* __restrict__ keys,
                                  const float* __restrict__ BX, const int* __restrict__ imidx,
                                  float* __restrict__ bk, float* __restrict__ topS,
                                  int* __restrict__ topC, int* __restrict__ topI,
                                  float* __restrict__ topOff, int* __restrict__ keep) {
    int i = blockIdx.x * blockDim.x + threadIdx.x;
    if (i >= K_NMS) return;
    unsigned long long kk = keys[i];
    unsigned u = (unsigned)(kk >> 32);
    unsigned idx = 0x7FFFFFFFu - (unsigned)(kk & 0xFFFFFFFFu);
    float s = funord(u);
    if (idx >= NBOX) { idx = 0; s = -1.0f; }
    int p = idx / NC, c = idx % NC;
    int im = imidx[p];
    bk[i * 4 + 0] = BX[idx * 4 + 0];
    bk[i * 4 + 1] = BX[idx * 4 + 1];
    bk[i * 4 + 2] = BX[idx * 4 + 2];
    bk[i * 4 + 3] = BX[idx * 4 + 3];
    topS[i] = s;
    topC[i] = c;
    topI[i] = im;
    topOff[i] = (float)(im * NC + c) * 1000.0f;
    keep[i] = (s > 0.0f) ? 1 : 0;
}

// ---------------- NMS (single block, sequential greedy) ----------------
__global__ void __launch_bounds__(1024)
nms_kernel(const float* __restrict__ bk, const float* __restrict__ topOff,
           int* __restrict__ keep) {
    __shared__ float sx1[K_NMS], sy1[K_NMS], sx2[K_NMS], sy2[K_NMS], sar[K_NMS];
    __shared__ int sK[1024];
    int tid = threadIdx.x;
    if (tid < K_NMS) {
        float off = topOff[tid];
        float a = bk[tid * 4 + 0] + off, b = bk[tid * 4 + 1] + off;
        float c = bk[tid * 4 + 2] + off, d = bk[tid * 4 + 3] + off;
        sx1[tid] = a; sy1[tid] = b; sx2[tid] = c; sy2[tid] = d;
        sar[tid] = (c - a) * (d - b);
        sK[tid] = keep[tid];
    } else {
        sK[tid] = 0;
    }
    for (int i = 0; i < K_NMS; ++i) {
        __syncthreads();
        if (sK[i]) {
            int j = tid;
            if (j > i && j < K_NMS && sK[j]) {
                float lx = fmaxf(sx1[i], sx1[j]);
                float ly = fmaxf(sy1[i], sy1[j]);
                float rx = fminf(sx2[i], sx2[j]);
                float ry = fminf(sy2[i], sy2[j]);
                float iw = fmaxf(rx - lx, 0.0f);
                float ih = fmaxf(ry - ly, 0.0f);
                float inter = iw * ih;
                float iou = inter / fmaxf(sar[i] + sar[j] - inter, 1e-9f);
                if (iou > 0.5f) sK[j] = 0;
            }
        }
    }
    __syncthreads();
    if (tid < K_NMS) keep[tid] = sK[tid];
}

// ---------------- per-image top-100 + output ----------------
__global__ void __launch_bounds__(1024)
output_kernel(const float* __restrict__ bk, const float* __restrict__ topS,
              const int* __restrict__ topC, const int* __restrict__ topI,
              const int* __restrict__ keep, float* __restrict__ out) {
    __shared__ unsigned long long sk[1024];
    int im = blockIdx.x;
    int tid = threadIdx.x;
    float s = -1.0f;
    if (tid < K_NMS && keep[tid] && topI[tid] == im) s = topS[tid];
    sk[tid] = ((unsigned long long)ford(s) << 32) |
              (unsigned long long)(unsigned)(1023 - tid);
    for (int k = 2; k <= 1024; k <<= 1) {
        for (int j = k >> 1; j > 0; j >>= 1) {
            __syncthreads();
            int ixj = tid ^ j;
            if (ixj > tid) {
                unsigned long long a = sk[tid], b = sk[ixj];
                bool up = ((tid & k) == 0);
                if (up ? (a < b) : (a > b)) { sk[tid] = b; sk[ixj] = a; }
            }
        }
    }
    __syncthreads();
    if (tid < 100) {
        unsigned long long kk = sk[tid];
        int j = 1023 - (int)(unsigned)(kk & 0xFFFFFFFFu);
        float sv = funord((unsigned)(kk >> 32));
        bool ok = sv > 0.0f;
        #pragma unroll
        for (int q = 0; q < 4; ++q) {
            float bv = 0.0f;
            if (ok) bv = bk[j * 4 + q];
            out[im * 400 + tid * 4 + q] = bv;
        }
        out[800 + im * 100 + tid]  = ok ? sv : 0.0f;
        out[1000 + im * 100 + tid] = ok ? (float)topC[j] : -1.0f;
    }
}

// ---------------- launcher ----------------
extern "C" void kernel_launch(void* const* d_in, const int* in_sizes, int n_in,
                              void* d_out, int out_size, void* d_ws, size_t ws_size,
                              hipStream_t stream) {
    const float* proposals = (const float*)d_in[0];
    const int*   imidx     = (const int*)  d_in[1];
    const float* fm0 = (const float*)d_in[2];
    const float* fm1 = (const float*)d_in[3];
    const float* fm2 = (const float*)d_in[4];
    const float* fm3 = (const float*)d_in[5];
    const float* W_fc0 = (const float*)d_in[6];
    const float* b_fc0 = (const float*)d_in[7];
    const float* W_fc1 = (const float*)d_in[8];
    const float* b_fc1 = (const float*)d_in[9];
    const float* W_cls = (const float*)d_in[10];
    const float* b_cls = (const float*)d_in[11];
    const float* W_reg = (const float*)d_in[12];
    const float* b_reg = (const float*)d_in[13];
    float* out = (float*)d_out;

    uintptr_t base = (uintptr_t)d_ws;
    size_t off = 0;
    auto alloc = [&](size_t bytes) -> void* {
        void* p = (void*)(base + off);
        off += (bytes + 255) & ~(size_t)255;
        return p;
    };
    float* A0     = (float*)alloc((size_t)M_PAD * FEAT * 4);   // roi features (padded)
    float* X1     = (float*)alloc((size_t)M_PAD * CLIN * 4);
    float* X2     = (float*)alloc((size_t)M_PAD * CLIN * 4);
    float* LOGB   = (float*)alloc((size_t)M_PAD * NCLS_P * 4);
    float* REGB   = (float*)alloc((size_t)M_PAD * 320 * 4);
    float* Wc_pad = (float*)alloc((size_t)CLIN * NCLS_P * 4);
    float* bc_pad = (float*)alloc((size_t)NCLS_P * 4);
    float* BX     = (float*)alloc((size_t)NBOX * 4 * 4);
    unsigned long long* KEYS = (unsigned long long*)alloc((size_t)SORTN * 8);
    float* bk     = (float*)alloc((size_t)K_NMS * 4 * 4);
    float* topS   = (float*)alloc((size_t)K_NMS * 4);
    int*   topC   = (int*)  alloc((size_t)K_NMS * 4);
    int*   topI   = (int*)  alloc((size_t)K_NMS * 4);
    float* topOff = (float*)alloc((size_t)K_NMS * 4);
    int*   keep   = (int*)  alloc((size_t)K_NMS * 4);

    // 1. zero pads + pad cls weights to 128 cols
    init_ws_kernel<<<((M_PAD - N_PROP) * FEAT + 255) / 256, 256, 0, stream>>>(
        A0, KEYS, W_cls, b_cls, Wc_pad, bc_pad);
    // 2. ROI align
    roi_align_kernel<<<N_PROP, 256, 0, stream>>>(proposals, imidx, fm0, fm1, fm2, fm3, A0);
    // 3. FC0 + ReLU  (1024 x 12544) @ (12544 x 1024)
    gemm_wmma_f32<<<dim3(16, 16), 256, 0, stream>>>(A0, W_fc0, b_fc0, X1, 1024, FEAT, 1);
    // 4. FC1 + ReLU
    gemm_wmma_f32<<<dim3(16, 16), 256, 0, stream>>>(X1, W_fc1, b_fc1, X2, 1024, CLIN, 1);
    // 5. cls head (padded to N=128)
    gemm_wmma_f32<<<dim3(2, 16), 256, 0, stream>>>(X2, Wc_pad, bc_pad, LOGB, NCLS_P, CLIN, 0);
    // 6. reg head (N=320)
    gemm_wmma_f32<<<dim3(5, 16), 256, 0, stream>>>(X2, W_reg, b_reg, REGB, 320, CLIN, 0);
    // 7. softmax + decode + sort keys
    decode_kernel<<<(N_PROP + 255) / 256, 256, 0, stream>>>(LOGB, REGB, proposals, BX, KEYS);
    // 8. global bitonic sort (descending) of 131072 keys
    for (int k = 2; k <= SORTN; k <<= 1)
        for (int j = k >> 1; j > 0; j >>= 1)
            bitonic_step_kernel<<<SORTN / 256, 256, 0, stream>>>(KEYS, k, j);
    // 9. gather top-1000
    gather_top_kernel<<<(K_NMS + 255) / 256, 256, 0, stream>>>(
        KEYS, BX, imidx, bk, topS, topC, topI, topOff, keep);
    // 10. NMS
    nms_kernel<<<1, 1024, 0, stream>>>(bk, topOff, keep);
    // 11. per-image top-100 outputs
    output_kernel<<<2, 1024, 0, stream>>>(bk, topS, topC, topI, keep, out);
}